// Model_21191368638832
// MI455X (gfx1250) — compile-verified
//
#include <hip/hip_runtime.h>
#include <hip/hip_bf16.h>

// ---------------------------------------------------------------------------
// Graph U-Net forward for MI455X (gfx1250).
// Dense GEMMs on V_WMMA_F32_16X16X4_F32 (fp32 matrix pipe, wave32).
// SAGEConv fused into one GEMM: A = [mean | self], B = [Wl ; Wr], K loop
// split into two branchless segments. OOB rows/cols are CLAMPED (never
// guarded) since A row m / B col n only feed C row m / col n, and stores are
// masked. All K are multiples of 4 (encoder zero-padded 3->4). N is a
// template parameter so B-fragment second-row loads use immediate offsets.
// All hot-path memory ops go through address_space(1) => global_load/store.
// ---------------------------------------------------------------------------

typedef float v2f __attribute__((ext_vector_type(2)));
typedef float v8f __attribute__((ext_vector_type(8)));

typedef const float __attribute__((address_space(1))) gcf;
typedef float       __attribute__((address_space(1))) gf;
typedef const v2f   __attribute__((address_space(1))) gcv2f;
typedef const int   __attribute__((address_space(1))) gci;

__device__ __forceinline__ float gelu_exact(float v) {
  return 0.5f * v * (1.0f + erff(v * 0.7071067811865475f));
}

// ---------------------------------------------------------------------------
// C[M,N] = act( A[M,K] @ [B0;B1] + bias[N] ), weight rows split at K1.
// Block: 256 threads = 8 waves arranged 4(M) x 2(N).
// Wave tile: 32 rows x 64 cols = 2 A-frags x 4 B-frags -> 8 WMMAs / 4-k-step.
// K, K1 multiples of 4. Fragment layouts per CDNA5 ISA 7.12.2 (f32, wave32).
// ---------------------------------------------------------------------------
template <int N>
__global__ __launch_bounds__(256)
void gemm_wmma_f32(const float* __restrict__ A,
                   const float* __restrict__ B0,
                   const float* __restrict__ B1,
                   const float* __restrict__ bias,
                   float* __restrict__ Cout,
                   int M, int K, int K1, int doGelu)
{
  const int lane    = threadIdx.x & 31;
  const int wave    = threadIdx.x >> 5;
  const int rowBase = blockIdx.x * 128 + (wave & 3) * 32;
  const int colBase = blockIdx.y * 128 + (wave >> 2) * 64;
  const int half    = lane >> 4;
  const int l15     = lane & 15;
  const int h2      = half * 2;

  v8f acc[2][4] = {};

  // Clamped per-lane A row pointers (garbage rows never stored).
  int ar0 = rowBase + l15;      if (ar0 > M - 1) ar0 = M - 1;
  int ar1 = rowBase + 16 + l15; if (ar1 > M - 1) ar1 = M - 1;
  gcf* Ap0 = (gcf*)(A + (long)ar0 * K + h2);
  gcf* Ap1 = (gcf*)(A + (long)ar1 * K + h2);

  // Clamped per-lane B column indices (garbage cols never stored).
  int cc[4];
#pragma unroll
  for (int s = 0; s < 4; ++s) {
    int c = colBase + s * 16 + l15;
    cc[s] = (c > N - 1) ? (N - 1) : c;
  }

  const float* Bseg[2] = {B0, B1};
  const int    kbeg[2] = {0, K1};
  const int    kend[2] = {K1, K};

#pragma unroll
  for (int seg = 0; seg < 2; ++seg) {
    const int iters = (kend[seg] - kbeg[seg]) >> 2;
    gcf* bp[4];
#pragma unroll
    for (int s = 0; s < 4; ++s) bp[s] = (gcf*)(Bseg[seg] + h2 * N + cc[s]);

    for (int it = 0; it < iters; ++it) {
      const v2f a0 = *(gcv2f*)Ap0;
      const v2f a1 = *(gcv2f*)Ap1;
      Ap0 += 4; Ap1 += 4;
      v2f b[4];
#pragma unroll
      for (int s = 0; s < 4; ++s) {
        b[s].x = bp[s][0];
        b[s].y = bp[s][N];          // immediate offset (N compile-time)
        bp[s] += 4 * N;             // constant stride bump
      }
#pragma unroll
      for (int s = 0; s < 4; ++s) {
        acc[0][s] = __builtin_amdgcn_wmma_f32_16x16x4_f32(
            false, a0, false, b[s], (short)0, acc[0][s], false, false);
        acc[1][s] = __builtin_amdgcn_wmma_f32_16x16x4_f32(
            false, a1, false, b[s], (short)0, acc[1][s], false, false);
      }
    }
  }

#pragma unroll
  for (int g = 0; g < 2; ++g) {
#pragma unroll
    for (int s = 0; s < 4; ++s) {
      const int col = colBase + s * 16 + l15;
      if (col >= N) continue;
      const float bv = ((gcf*)bias)[col];
#pragma unroll
      for (int r = 0; r < 8; ++r) {
        const int row = rowBase + g * 16 + half * 8 + r;
        if (row < M) {
          float v = acc[g][s][r] + bv;
          if (doGelu) v = gelu_exact(v);
          ((gf*)Cout)[(long)row * N + col] = v;
        }
      }
    }
  }
}

// ---------------------------------------------------------------------------
// Ahat[r, 0:C]  = (1/16) * sum_{j<16} Z[src[16r+j], :]   (fixed in-degree 16)
// Ahat[r, C:2C] = Z[r, :]
// ---------------------------------------------------------------------------
__global__ __launch_bounds__(256)
void aggregate16(const float* __restrict__ Z, const int* __restrict__ src,
                 float* __restrict__ Ahat, int n, int C)
{
  const int gwave = (int)((blockIdx.x * (long)blockDim.x + threadIdx.x) >> 5);
  const int lane  = threadIdx.x & 31;
  if (gwave >= n) return;
  const int r = gwave;
  gcf* Zg = (gcf*)Z;
  gf*  Ag = (gf*)Ahat;
  const int srcv = (lane < 16) ? ((gci*)src)[r * 16 + lane] : 0;
  const long base = (long)r * (long)(2 * C);
  for (int c = lane; c < C; c += 32) {
    float s = 0.0f;
#pragma unroll
    for (int j = 0; j < 16; ++j) {
      const int sj = __shfl(srcv, j, 32);
      s += Zg[(long)sj * C + c];
    }
    Ag[base + c]     = s * 0.0625f;
    Ag[base + C + c] = Zg[(long)r * C + c];
  }
}

// dst[r, dstOff + c] = src[(idx ? idx[r] : r), c]
__global__ void gather_rows(const float* __restrict__ src, const int* __restrict__ idx,
                            float* __restrict__ dst, int n, int C,
                            int dstStride, int dstOff)
{
  gcf* sg = (gcf*)src;
  gf*  dg = (gf*)dst;
  const long total = (long)n * (long)C;
  for (long t = blockIdx.x * (long)blockDim.x + threadIdx.x; t < total;
       t += (long)gridDim.x * blockDim.x) {
    const int r = (int)(t / C);
    const int c = (int)(t % C);
    const int s = idx ? ((gci*)idx)[r] : r;
    dg[(long)r * dstStride + dstOff + c] = sg[(long)s * C + c];
  }
}

// zero-pad encoder inputs: xpad[n,4] from x[n,3]; w1pad[4,128] from W1[3,128]
__global__ void pad_encoder(const float* __restrict__ x, const float* __restrict__ W1,
                            float* __restrict__ xpad, float* __restrict__ w1pad, int n)
{
  const long t0 = blockIdx.x * (long)blockDim.x + threadIdx.x;
  const long totX = (long)n * 4;
  for (long i = t0; i < totX; i += (long)gridDim.x * blockDim.x) {
    const int r = (int)(i >> 2), c = (int)(i & 3);
    ((gf*)xpad)[i] = (c < 3) ? ((gcf*)x)[r * 3 + c] : 0.0f;
  }
  if (t0 < 512) ((gf*)w1pad)[t0] = (t0 < 384) ? ((gcf*)W1)[t0] : 0.0f;
}

// Per-channel sum / sum-of-squares; block = 64 cols x 4 row groups.
__global__ __launch_bounds__(256)
void bn_stats(const float* __restrict__ Z, int n, int C, float* __restrict__ accum)
{
  const int col = blockIdx.x * 64 + (threadIdx.x & 63);
  const int rg  = threadIdx.x >> 6;
  gcf* Zg = (gcf*)Z;
  float s = 0.0f, s2 = 0.0f;
  if (col < C) {
    for (int r = blockIdx.y * 4 + rg; r < n; r += gridDim.y * 4) {
      const float v = Zg[(long)r * C + col];
      s += v; s2 += v * v;
    }
  }
  __shared__ float ls[256], ls2[256];
  ls[threadIdx.x] = s; ls2[threadIdx.x] = s2;
  __syncthreads();
  if (rg == 0 && col < C) {
    const int t = threadIdx.x;
    s  = ls[t]  + ls[t + 64]  + ls[t + 128]  + ls[t + 192];
    s2 = ls2[t] + ls2[t + 64] + ls2[t + 128] + ls2[t + 192];
    atomicAdd(&accum[col], s);
    atomicAdd(&accum[C + col], s2);
  }
}

__global__ void bn_finalize(const float* __restrict__ accum,
                            const float* __restrict__ g, const float* __restrict__ b,
                            float* __restrict__ ss, int n, int C)
{
  const int c = blockIdx.x * blockDim.x + threadIdx.x;
  if (c >= C) return;
  const float inv_n = 1.0f / (float)n;
  const float mean = accum[c] * inv_n;
  const float var  = accum[C + c] * inv_n - mean * mean;  // biased
  const float sc   = g[c] * rsqrtf(var + 1e-5f);
  ss[c]     = sc;
  ss[C + c] = b[c] - mean * sc;
}

__global__ void bn_apply(float* __restrict__ Z, const float* __restrict__ ss,
                         int n, int C, int doGelu)
{
  gf* Zg = (gf*)Z;
  gcf* sg = (gcf*)ss;
  const long total = (long)n * (long)C;
  for (long t = blockIdx.x * (long)blockDim.x + threadIdx.x; t < total;
       t += (long)gridDim.x * blockDim.x) {
    const int c = (int)(t % C);
    float v = Zg[t] * sg[c] + sg[C + c];
    if (doGelu) v = gelu_exact(v);
    Zg[t] = v;
  }
}

// ---------------------------------------------------------------------------
static inline int ceil_div_i(int a, int b) { return (a + b - 1) / b; }
static inline int grid_for(long total, int block, int cap) {
  long g = (total + block - 1) / block;
  if (g > cap) g = cap;
  if (g < 1) g = 1;
  return (int)g;
}

extern "C" void kernel_launch(void* const* d_in, const int* in_sizes, int n_in,
                              void* d_out, int out_size, void* d_ws, size_t ws_size,
                              hipStream_t stream)
{
  (void)in_sizes; (void)n_in; (void)out_size; (void)ws_size;

  // ---- inputs (setup_inputs dict leaf order) ----
  const float* x = (const float*)d_in[0];
  const int* geoS = (const int*)d_in[1];                       // [2,E] -> src first
  const int* id_[4]   = {(const int*)d_in[2], (const int*)d_in[3],
                         (const int*)d_in[4], (const int*)d_in[5]};
  const int* edgeS[5] = {geoS, (const int*)d_in[6], (const int*)d_in[7],
                         (const int*)d_in[8], (const int*)d_in[9]};
  const int* clus[5]  = {nullptr, (const int*)d_in[10], (const int*)d_in[11],
                         (const int*)d_in[12], (const int*)d_in[13]};
  const float* encW1 = (const float*)d_in[14];
  const float* encb1 = (const float*)d_in[15];
  const float* encW2 = (const float*)d_in[16];
  const float* encb2 = (const float*)d_in[17];
  const float* decW1 = (const float*)d_in[18];
  const float* decb1 = (const float*)d_in[19];
  const float* decW2 = (const float*)d_in[20];
  const float* decb2 = (const float*)d_in[21];
  const float *dWl[5], *dbl[5], *dWr[5];
  for (int l = 0; l < 5; ++l) {
    dWl[l] = (const float*)d_in[22 + 3 * l];
    dbl[l] = (const float*)d_in[23 + 3 * l];
    dWr[l] = (const float*)d_in[24 + 3 * l];
  }
  const float *uWl[4], *ubl[4], *uWr[4];
  for (int j = 0; j < 4; ++j) {
    uWl[j] = (const float*)d_in[37 + 3 * j];
    ubl[j] = (const float*)d_in[38 + 3 * j];
    uWr[j] = (const float*)d_in[39 + 3 * j];
  }
  const float *bng[9], *bnb[9];
  for (int k = 0; k < 9; ++k) {
    bng[k] = (const float*)d_in[49 + 2 * k];
    bnb[k] = (const float*)d_in[50 + 2 * k];
  }

  // ---- model dims ----
  const int NS[5]   = {80000, 40000, 20000, 10000, 5000};
  const int inD[5]  = {64, 64, 128, 256, 512};
  const int outD[5] = {64, 128, 256, 512, 1024};

  // ---- workspace layout (floats) ----
  float* ws = (float*)d_ws;
  float* bufA  = ws;                       // 30,720,000 (max Ahat: e.g. 80000x384)
  float* bufB  = ws + 30720000;            // 15,360,000 (max zcat: 80000x192)
  float* bufC  = ws + 46080000;            //  5,120,000 (up outputs; also xpad)
  float* zl[5];
  for (int l = 0; l < 5; ++l) zl[l] = ws + 51200000 + (size_t)l * 5120000;
  float* bnAcc = ws + 76800000;            // 2048
  float* bnSS  = ws + 76802048;            // 2048
  float* w1pad = ws + 76804096;            // 512

  auto gemm = [&](const float* A, const float* B0, const float* B1,
                  const float* bias, float* C, int M, int N, int K, int K1,
                  int gelu) {
    dim3 g(ceil_div_i(M, 128), ceil_div_i(N, 128));
    switch (N) {
      case 4:
        gemm_wmma_f32<4><<<g, 256, 0, stream>>>(A, B0, B1, bias, C, M, K, K1, gelu);
        break;
      case 64:
        gemm_wmma_f32<64><<<g, 256, 0, stream>>>(A, B0, B1, bias, C, M, K, K1, gelu);
        break;
      case 128:
        gemm_wmma_f32<128><<<g, 256, 0, stream>>>(A, B0, B1, bias, C, M, K, K1, gelu);
        break;
      case 256:
        gemm_wmma_f32<256><<<g, 256, 0, stream>>>(A, B0, B1, bias, C, M, K, K1, gelu);
        break;
      case 512:
        gemm_wmma_f32<512><<<g, 256, 0, stream>>>(A, B0, B1, bias, C, M, K, K1, gelu);
        break;
      case 1024:
        gemm_wmma_f32<1024><<<g, 256, 0, stream>>>(A, B0, B1, bias, C, M, K, K1, gelu);
        break;
      default: break;  // unreachable for this model
    }
  };
  auto gather = [&](const float* src, const int* idx, float* dst, int n, int C,
                    int dstStride, int dstOff) {
    gather_rows<<<grid_for((long)n * C, 256, 8192), 256, 0, stream>>>(
        src, idx, dst, n, C, dstStride, dstOff);
  };
  auto aggregate = [&](const float* Z, const int* src, float* Ahat, int n, int C) {
    aggregate16<<<ceil_div_i(n, 8), 256, 0, stream>>>(Z, src, Ahat, n, C);
  };
  auto run_bn = [&](float* Z, int n, int C, const float* g, const float* b,
                    int gelu) {
    hipMemsetAsync(bnAcc, 0, 2 * (size_t)C * sizeof(float), stream);
    dim3 gs(ceil_div_i(C, 64), 64);
    bn_stats<<<gs, 256, 0, stream>>>(Z, n, C, bnAcc);
    bn_finalize<<<ceil_div_i(C, 256), 256, 0, stream>>>(bnAcc, g, b, bnSS, n, C);
    bn_apply<<<grid_for((long)n * C, 256, 8192), 256, 0, stream>>>(Z, bnSS, n, C, gelu);
  };

  // ---- encoder MLP: pad K 3->4, then gelu(x@W1+b1) -> @W2+b2 ----
  pad_encoder<<<grid_for((long)NS[0] * 4, 256, 8192), 256, 0, stream>>>(
      x, encW1, bufC, w1pad, NS[0]);
  gemm(bufC, w1pad, w1pad, encb1, bufA, NS[0], 128, 4,   4,   1);
  gemm(bufA, encW2, encW2, encb2, bufB, NS[0], 64,  128, 128, 0);

  // ---- down path ----
  for (int l = 0; l < 5; ++l) {
    const float* zin = bufB;
    if (l > 0) {
      gather(zl[l - 1], id_[l - 1], bufB, NS[l], inD[l], inD[l], 0);
    }
    aggregate(zin, edgeS[l], bufA, NS[l], inD[l]);
    gemm(bufA, dWl[l], dWr[l], dbl[l], zl[l],
         NS[l], outD[l], 2 * inD[l], inD[l], 0);
    run_bn(zl[l], NS[l], outD[l], bng[l], bnb[l], 1);
  }

  // ---- up path (n = 4..1) ----
  for (int n = 4; n >= 1; --n) {
    const int iup   = outD[n];
    const int iskip = outD[n - 1];
    const int icat  = iup + iskip;
    const int nn    = NS[n - 1];
    const float* zsrc = (n == 4) ? zl[4] : bufC;
    gather(zsrc,      clus[n], bufB, nn, iup,   icat, 0);     // z[clus]
    gather(zl[n - 1], nullptr, bufB, nn, iskip, icat, iup);   // concat skip
    aggregate(bufB, edgeS[n - 1], bufA, nn, icat);
    gemm(bufA, uWl[n - 1], uWr[n - 1], ubl[n - 1], bufC,
         nn, outD[n - 1], 2 * icat, icat, 0);
    run_bn(bufC, nn, outD[n - 1], bng[4 + n], bnb[4 + n], (n != 1) ? 1 : 0);
  }

  // ---- decoder MLP ----
  gemm(bufC, decW1, decW1, decb1, bufA, NS[0], 128, 64, 64, 1);
  gemm(bufA, decW2, decW2, decb2, (float*)d_out, NS[0], 4, 128, 128, 0);
}